// DynFilter_68891275428414
// MI455X (gfx1250) — compile-verified
//
#include <hip/hip_runtime.h>
#include <hip/hip_bf16.h>
#include <stdint.h>

typedef float v4f __attribute__((ext_vector_type(4)));

#define B_ 4
#define C_ 3
#define H_ 512
#define W_ 512
#define KH_ 5
#define KW_ 5
#define PAD_ 2
#define TAPS (C_ * KH_ * KW_)          // 75
#define HW_ ((size_t)H_ * (size_t)W_)  // 262144

// Block tile: 128 px wide x 8 px tall; block = (32,8); each thread -> 4 px.
#define TILE_W 128
#define TILE_H 8
#define LDS_W  (TILE_W + 8)            // cols map to gx = tx0-4+col : [tx0-4, tx0+131], 16B aligned
#define LDS_H  (TILE_H + 4)            // +/-2 halo rows -> 12
#define LDS_ROWS   (C_ * LDS_H)        // 36 rows of 136 floats
#define LDS_QUADS_PER_ROW (LDS_W / 4)  // 34 float4 quads per row
#define LDS_QUADS  (LDS_ROWS * LDS_QUADS_PER_ROW) // 1224 b128 transfers
#define LDS_ELEMS  (C_ * LDS_H * LDS_W)           // 4896 floats = 19.1 KB

// CDNA5 async global->LDS copy (b128): VDST = per-lane LDS byte address, VADDR = 64-bit global addr.
__device__ __forceinline__ void async_ld_b128(unsigned lds_byte_addr, const float* gaddr) {
    asm volatile("global_load_async_to_lds_b128 %0, %1, off"
                 :: "v"(lds_byte_addr), "v"(gaddr)
                 : "memory");
}
__device__ __forceinline__ void wait_asynccnt0() {
    asm volatile("s_wait_asynccnt 0" ::: "memory");
}

__global__ __launch_bounds__(256) void DynFilter_68891275428414_kernel(
    const float* __restrict__ x,
    const float* __restrict__ filt,
    float* __restrict__ out) {

    __shared__ float smem[LDS_ELEMS];

    const int tx0 = blockIdx.x * TILE_W;
    const int ty0 = blockIdx.y * TILE_H;
    const int b   = blockIdx.z;

    // ---------------- Stage x tile (with zero-padded halo) into LDS ---------------------
    // Every 16B quad is either fully in-bounds or fully out-of-bounds:
    //   x: gx = tx0-4+4k, W and tx0 are multiples of 128 -> quad straddles nothing.
    //   y: per-row property trivially.
    // In-bounds quads: one async b128 global->LDS transfer. OOB quads: b128 zero store.
    const int lin = threadIdx.y * 32 + threadIdx.x; // 0..255
    for (int e = lin; e < LDS_QUADS; e += 256) {
        const int c   = e / (LDS_H * LDS_QUADS_PER_ROW);
        const int rem = e - c * (LDS_H * LDS_QUADS_PER_ROW);
        const int r   = rem / LDS_QUADS_PER_ROW;
        const int k   = rem - r * LDS_QUADS_PER_ROW;
        const int gy  = ty0 - PAD_ + r;
        const int gx  = tx0 - 4 + k * 4;
        float* lptr = &smem[(c * LDS_H + r) * LDS_W + k * 4];
        if ((unsigned)gy < (unsigned)H_ && (unsigned)gx < (unsigned)W_) {
            const float* g = x + (((size_t)(b * C_ + c) * H_ + (size_t)gy) * W_ + (size_t)gx);
            async_ld_b128((unsigned)(uintptr_t)lptr, g); // low 32 bits of flat LDS addr == DS byte addr
        } else {
            *(v4f*)lptr = (v4f){0.f, 0.f, 0.f, 0.f};     // zero padding for im2col edges
        }
    }
    wait_asynccnt0();
    __syncthreads();

    // ---------------- 75-tap dynamic-filter reduction -----------------------------------
    const int y   = ty0 + threadIdx.y;         // output row
    const int px0 = tx0 + threadIdx.x * 4;     // first of 4 output pixels (16B aligned)

    // filter[b, t, y, px0..px0+3]; tap planes are HW_ apart
    const float* fptr = filt + (size_t)b * TAPS * HW_ + (size_t)y * W_ + (size_t)px0;

    v4f acc = {0.f, 0.f, 0.f, 0.f};

#pragma unroll
    for (int c = 0; c < C_; ++c) {
#pragma unroll
        for (int i = 0; i < KH_; ++i) {
            // LDS row for this (c,i): global row (y + i - 2) -> index (threadIdx.y + i)
            const float* lrow =
                &smem[(c * LDS_H + (threadIdx.y + i)) * LDS_W + threadIdx.x * 4];
            // 12-float window covering x columns [px0-4, px0+7] (3 aligned b128 LDS loads)
            const v4f a0 = *(const v4f*)(lrow);
            const v4f a1 = *(const v4f*)(lrow + 4);
            const v4f a2 = *(const v4f*)(lrow + 8);
            float w[12];
            w[0] = a0[0]; w[1]  = a0[1]; w[2]  = a0[2]; w[3]  = a0[3];
            w[4] = a1[0]; w[5]  = a1[1]; w[6]  = a1[2]; w[7]  = a1[3];
            w[8] = a2[0]; w[9]  = a2[1]; w[10] = a2[2]; w[11] = a2[3];
#pragma unroll
            for (int j = 0; j < KW_; ++j) {
                const int t = c * (KH_ * KW_) + i * KW_ + j;
                // read-once stream -> non-temporal load (keep 192MB L2 for x reuse)
                const v4f f = __builtin_nontemporal_load((const v4f*)(fptr + (size_t)t * HW_));
                // pixel p uses x[y+i-2, px0+p+j-2] == w[p + j + 2]
                acc[0] = fmaf(w[j + 2], f[0], acc[0]);
                acc[1] = fmaf(w[j + 3], f[1], acc[1]);
                acc[2] = fmaf(w[j + 4], f[2], acc[2]);
                acc[3] = fmaf(w[j + 5], f[3], acc[3]);
            }
        }
    }

    // out[b, 0, y, px0..px0+3] — write-once -> non-temporal store
    float* op = out + ((size_t)b * H_ + (size_t)y) * W_ + (size_t)px0;
    __builtin_nontemporal_store(acc, (v4f*)op);
}

extern "C" void kernel_launch(void* const* d_in, const int* in_sizes, int n_in,
                              void* d_out, int out_size, void* d_ws, size_t ws_size,
                              hipStream_t stream) {
    const float* x    = (const float*)d_in[0];
    const float* filt = (const float*)d_in[1];
    float* out        = (float*)d_out;

    dim3 block(32, 8, 1);                       // 256 threads = 8 wave32
    dim3 grid(W_ / TILE_W, H_ / TILE_H, B_);    // (4, 64, 4)
    hipLaunchKernelGGL(DynFilter_68891275428414_kernel, grid, block, 0, stream,
                       x, filt, out);
}